// ContinuousThoughtMachine_8839042695343
// MI455X (gfx1250) — compile-verified
//
#include <hip/hip_runtime.h>
#include <hip/hip_bf16.h>
#include <math.h>

typedef __attribute__((ext_vector_type(2))) float v2f;
typedef __attribute__((ext_vector_type(8))) float v8f;

#define ITERS 50
#define BATCH 64
#define DDIM  2048
#define EDIM  512
#define ODIM  4096
#define MWIN  25

// ---------------- reductions ----------------
__device__ __forceinline__ float waveRedSum(float v){
#pragma unroll
  for (int off = 16; off; off >>= 1) v += __shfl_xor(v, off, 32);
  return v;
}
__device__ __forceinline__ float waveRedMax(float v){
#pragma unroll
  for (int off = 16; off; off >>= 1) v = fmaxf(v, __shfl_xor(v, off, 32));
  return v;
}
__device__ __forceinline__ float blockRedSum(float v, float* sh){
  int lane = threadIdx.x & 31, w = threadIdx.x >> 5;
  v = waveRedSum(v);
  if (lane == 0) sh[w] = v;
  __syncthreads();
  if (w == 0){
    float x = (lane < (int)(blockDim.x >> 5)) ? sh[lane] : 0.f;
    x = waveRedSum(x);
    if (lane == 0) sh[0] = x;
  }
  __syncthreads();
  float r = sh[0];
  __syncthreads();
  return r;
}
__device__ __forceinline__ float blockRedMax(float v, float* sh){
  int lane = threadIdx.x & 31, w = threadIdx.x >> 5;
  v = waveRedMax(v);
  if (lane == 0) sh[w] = v;
  __syncthreads();
  if (w == 0){
    float x = (lane < (int)(blockDim.x >> 5)) ? sh[lane] : -3.4e38f;
    x = waveRedMax(x);
    if (lane == 0) sh[0] = x;
  }
  __syncthreads();
  float r = sh[0];
  __syncthreads();
  return r;
}
__device__ __forceinline__ float sigmoidf(float x){ return 1.f / (1.f + expf(-x)); }

// ---------------- generic f32 WMMA GEMM: C(MxN) = A(MxK) @ B(KxN) + bias ----------------
// One wave computes a 16x64 tile of C (4 WMMA accumulators along N). 8 waves/block.
// Requires M%16==0, N%64==0, K%4==0.
__global__ void gemm_wmma_f32(const float* __restrict__ A, int lda,
                              const float* __restrict__ B, int ldb,
                              float* __restrict__ C, int ldc,
                              const float* __restrict__ bias,
                              int M, int N, int K){
  int wave = threadIdx.x >> 5, lane = threadIdx.x & 31;
  int lm = lane & 15, half = lane >> 4;
  int n64 = N >> 6;
  int gid = blockIdx.x * 8 + wave;
  int total = (M >> 4) * n64;
  if (gid >= total) return;                 // wave-uniform
  int mt = gid / n64, nt = gid - mt * n64;
  int m0 = mt << 4, n0 = nt << 6;
  v8f acc0 = {}, acc1 = {}, acc2 = {}, acc3 = {};
  const float* arow = A + (size_t)(m0 + lm) * lda;
  const float* bbase = B + n0 + lm;
  for (int kk = 0; kk < K; kk += 4){
    int ka = kk + (half << 1);
    v2f a; a.x = arow[ka]; a.y = arow[ka + 1];
    const float* bp = bbase + (size_t)ka * ldb;
    v2f b0; b0.x = bp[0];  b0.y = bp[ldb];
    v2f b1; b1.x = bp[16]; b1.y = bp[ldb + 16];
    v2f b2; b2.x = bp[32]; b2.y = bp[ldb + 32];
    v2f b3; b3.x = bp[48]; b3.y = bp[ldb + 48];
    acc0 = __builtin_amdgcn_wmma_f32_16x16x4_f32(false, a, false, b0, (short)0, acc0, false, false);
    acc1 = __builtin_amdgcn_wmma_f32_16x16x4_f32(false, a, false, b1, (short)0, acc1, false, false);
    acc2 = __builtin_amdgcn_wmma_f32_16x16x4_f32(false, a, false, b2, (short)0, acc2, false, false);
    acc3 = __builtin_amdgcn_wmma_f32_16x16x4_f32(false, a, false, b3, (short)0, acc3, false, false);
  }
  float bi0 = 0.f, bi1 = 0.f, bi2 = 0.f, bi3 = 0.f;
  if (bias){ bi0 = bias[n0+lm]; bi1 = bias[n0+16+lm]; bi2 = bias[n0+32+lm]; bi3 = bias[n0+48+lm]; }
#pragma unroll
  for (int v = 0; v < 8; v++){
    int r = m0 + v + (half << 3);
    float* crow = C + (size_t)r * ldc + n0 + lm;
    crow[0]  = acc0[v] + bi0;
    crow[16] = acc1[v] + bi1;
    crow[32] = acc2[v] + bi2;
    crow[48] = acc3[v] + bi3;
  }
}

// ---------------- feats: x(B,12,64) -> feats(B*64, 12) ----------------
__global__ void feats_kernel(const float* __restrict__ x, float* __restrict__ feats){
  int idx = blockIdx.x * 256 + threadIdx.x;     // 4096*12
  if (idx >= 4096 * 12) return;
  int row = idx / 12, c = idx - row * 12;
  int b = row >> 6, s = row & 63;
  feats[idx] = x[b * 768 + c * 64 + s];
}

// ---------------- in-place LayerNorm over rows of length 512 ----------------
__global__ void ln512_kernel(float* __restrict__ buf, const float* __restrict__ g,
                             const float* __restrict__ be){
  __shared__ float red[8];
  int r = blockIdx.x;
  float* row = buf + (size_t)r * 512;
  int d0 = threadIdx.x, d1 = threadIdx.x + 256;
  float v0 = row[d0], v1 = row[d1];
  float sum   = blockRedSum(v0 + v1, red);
  float sumsq = blockRedSum(v0 * v0 + v1 * v1, red);
  float mean = sum * (1.f / 512.f);
  float var  = sumsq * (1.f / 512.f) - mean * mean;
  float inv = rsqrtf(var + 1e-5f);
  row[d0] = (v0 - mean) * inv * g[d0] + be[d0];
  row[d1] = (v1 - mean) * inv * g[d1] + be[d1];
}

// ---------------- init per-call state ----------------
__global__ void init_kernel(const float* __restrict__ start_act,
                            float* aA, float* bA, float* aO, float* bO,
                            float* __restrict__ act, float* __restrict__ pre){
  int idx = blockIdx.x * 256 + threadIdx.x;   // 131072
  if (idx < 32768){ aA[idx] = 0.f; bA[idx] = 0.f; aO[idx] = 0.f; bO[idx] = 0.f; }
  if (idx < BATCH * DDIM){
    int b = idx >> 11, d = idx & 2047;
    float v = start_act[d];
    act[idx] = v;
    pre[(size_t)b * 2560 + 512 + d] = v;
  }
}

// ---------------- synchronisation (shared by action/out paths) ----------------
__global__ void sync_kernel(const float* __restrict__ act,
                            const int* __restrict__ il, const int* __restrict__ ir,
                            const float* __restrict__ decay,
                            float* __restrict__ a_s, float* __restrict__ b_s,
                            float* __restrict__ out){
  int idx = blockIdx.x * 256 + threadIdx.x;    // 64*512
  if (idx >= BATCH * 512) return;
  int b = idx >> 9, i = idx & 511;
  float dc = fminf(fmaxf(decay[i], 0.f), 15.f);
  float r = expf(-dc);
  const float* arow = act + (size_t)b * DDIM;
  float pw = arow[il[i]] * arow[ir[i]];
  float a = r * a_s[idx] + pw;
  float bb = r * b_s[idx] + 1.f;
  a_s[idx] = a; b_s[idx] = bb;
  out[idx] = a * rsqrtf(bb);
}

// ---------------- single-query MHA over 64 tokens; one wave per (b,h) ----------------
__global__ void attn_kernel(const float* __restrict__ qh, const float* __restrict__ kp,
                            const float* __restrict__ vp, float* __restrict__ o){
  __shared__ float ash[8][64];
  int wave = threadIdx.x >> 5, lane = threadIdx.x & 31;
  int idx = blockIdx.x * 8 + wave;        // 0..511 = b*8 + h
  int b = idx >> 3, h = idx & 7;
  const float* q = qh + (size_t)b * 512 + h * 64;
  const float* kbase = kp + (size_t)b * 64 * 512 + h * 64;
  float s0 = 0.f, s1 = 0.f;
  for (int d = 0; d < 64; d++){
    float qd = q[d];
    s0 += qd * kbase[(size_t)lane * 512 + d];
    s1 += qd * kbase[(size_t)(lane + 32) * 512 + d];
  }
  s0 *= 0.125f; s1 *= 0.125f;
  float m = waveRedMax(fmaxf(s0, s1));
  float e0 = expf(s0 - m), e1 = expf(s1 - m);
  float inv = 1.f / waveRedSum(e0 + e1);
  ash[wave][lane] = e0 * inv; ash[wave][lane + 32] = e1 * inv;
  __syncthreads();
  const float* vbase = vp + (size_t)b * 64 * 512 + h * 64;
  float o0 = 0.f, o1 = 0.f;
  for (int s = 0; s < 64; s++){
    float a = ash[wave][s];
    o0 += a * vbase[(size_t)s * 512 + lane];
    o1 += a * vbase[(size_t)s * 512 + lane + 32];
  }
  float* orow = o + (size_t)b * 512 + h * 64;
  orow[lane] = o0; orow[lane + 32] = o1;
}

// ---------------- GLU + LayerNorm(D=2048); write state into ring slot t%25 ----------------
__global__ void glu_ln_kernel(const float* __restrict__ y, const float* __restrict__ g,
                              const float* __restrict__ be, float* __restrict__ ring, int t){
  __shared__ float sh[2048];
  __shared__ float red[8];
  int b = blockIdx.x;
  const float* row = y + (size_t)b * 4096;
  float s = 0.f, ss = 0.f;
#pragma unroll
  for (int i = 0; i < 8; i++){
    int d = threadIdx.x + i * 256;
    float v = row[d] * sigmoidf(row[2048 + d]);
    sh[d] = v; s += v; ss += v * v;
  }
  float sum   = blockRedSum(s, red);
  float sumsq = blockRedSum(ss, red);
  float mean = sum * (1.f / 2048.f);
  float var  = sumsq * (1.f / 2048.f) - mean * mean;
  float inv = rsqrtf(var + 1e-5f);
  float* dst = ring + (size_t)(t % MWIN) * (BATCH * DDIM) + (size_t)b * DDIM;
#pragma unroll
  for (int i = 0; i < 8; i++){
    int d = threadIdx.x + i * 256;
    dst[d] = (sh[d] - mean) * inv * g[d] + be[d];
  }
}

// ---------------- per-neuron NLMs: trace(64x25) @ w1(25x64) via WMMA, then GLU + tiny NLM2 ----------------
__device__ __forceinline__ float trace_val(const float* __restrict__ ring,
                                           const float* __restrict__ st,
                                           int t, int b, int n, int m){
  if (m >= MWIN) return 0.f;                        // K padded 25 -> 28
  int sidx = t - (24 - m);
  if (sidx >= 0)
    return ring[(size_t)(sidx % MWIN) * (BATCH * DDIM) + (size_t)b * DDIM + n];
  return st[(size_t)n * MWIN + (MWIN + sidx)];      // start_trace[d=n, col]
}

__global__ void nlm_kernel(const float* __restrict__ ring, const float* __restrict__ st,
                           const float* __restrict__ w1, const float* __restrict__ b1,
                           const float* __restrict__ w2, const float* __restrict__ b2,
                           float* __restrict__ act, float* __restrict__ pre, int t){
  __shared__ float hbuf[8][512];                    // per wave: 16 rows x 32 glu outputs
  int wave = threadIdx.x >> 5, lane = threadIdx.x & 31;
  int lm = lane & 15, half = lane >> 4;
  int n = blockIdx.x * 2 + (wave >> 2);             // neuron id (0..2047)
  int b0 = (wave & 3) << 4;                         // 16-row batch block
  int b = b0 + lm;
  v8f acc0 = {}, acc1 = {}, acc2 = {}, acc3 = {};
  for (int kk = 0; kk < 28; kk += 4){
    int ka = kk + (half << 1);
    v2f a;
    a.x = trace_val(ring, st, t, b, n, ka);
    a.y = trace_val(ring, st, t, b, n, ka + 1);
    bool ok0 = ka < MWIN, ok1 = (ka + 1) < MWIN;
    const float* wp = w1 + (size_t)ka * 131072 + (size_t)lm * 2048 + n;  // w1[m][h][n]
    v2f b0v, b1v, b2v, b3v;
    b0v.x = ok0 ? wp[0]        : 0.f;  b0v.y = ok1 ? wp[131072]           : 0.f;
    b1v.x = ok0 ? wp[32768]    : 0.f;  b1v.y = ok1 ? wp[131072 + 32768]   : 0.f;
    b2v.x = ok0 ? wp[65536]    : 0.f;  b2v.y = ok1 ? wp[131072 + 65536]   : 0.f;
    b3v.x = ok0 ? wp[98304]    : 0.f;  b3v.y = ok1 ? wp[131072 + 98304]   : 0.f;
    acc0 = __builtin_amdgcn_wmma_f32_16x16x4_f32(false, a, false, b0v, (short)0, acc0, false, false);
    acc1 = __builtin_amdgcn_wmma_f32_16x16x4_f32(false, a, false, b1v, (short)0, acc1, false, false);
    acc2 = __builtin_amdgcn_wmma_f32_16x16x4_f32(false, a, false, b2v, (short)0, acc2, false, false);
    acc3 = __builtin_amdgcn_wmma_f32_16x16x4_f32(false, a, false, b3v, (short)0, acc3, false, false);
  }
  float bb0 = b1[n * 64 + lm],      bb1 = b1[n * 64 + 16 + lm];
  float bb2 = b1[n * 64 + 32 + lm], bb3 = b1[n * 64 + 48 + lm];
#pragma unroll
  for (int v = 0; v < 8; v++){
    int rl = v + (half << 3);
    float p0 = acc0[v] + bb0, p2 = acc2[v] + bb2;   // GLU pair (h, h+32)
    float p1 = acc1[v] + bb1, p3 = acc3[v] + bb3;
    hbuf[wave][rl * 32 + lm]      = p0 * sigmoidf(p2);
    hbuf[wave][rl * 32 + 16 + lm] = p1 * sigmoidf(p3);
  }
  __syncthreads();
  if (lane < 16){
    float a0 = b2[n * 2], a1 = b2[n * 2 + 1];
    const float* w2p = w2 + n;                      // w2[j][c][n], stride j = 4096
    for (int j = 0; j < 32; j++){
      float hv = hbuf[wave][lane * 32 + j];
      a0 += hv * w2p[(size_t)j * 4096];
      a1 += hv * w2p[(size_t)j * 4096 + 2048];
    }
    float av = a0 * sigmoidf(a1);
    int bb = b0 + lane;
    act[(size_t)bb * DDIM + n] = av;
    pre[(size_t)bb * 2560 + 512 + n] = av;
  }
}

// ---------------- write predictions (strided) + certainties ----------------
__global__ void finalize_kernel(const float* __restrict__ pred, float* __restrict__ d_out, int t){
  __shared__ float red[8];
  int b = blockIdx.x;
  const float* row = pred + (size_t)b * ODIM;
  float m = -3.4e38f;
  for (int i = threadIdx.x; i < ODIM; i += 256) m = fmaxf(m, row[i]);
  m = blockRedMax(m, red);
  float s = 0.f;
  for (int i = threadIdx.x; i < ODIM; i += 256) s += expf(row[i] - m);
  s = blockRedSum(s, red);
  float lse = m + logf(s);
  float acc = 0.f;
  float* pout = d_out + (size_t)b * ODIM * ITERS + t;
  for (int i = threadIdx.x; i < ODIM; i += 256){
    float x = row[i];
    float lp = x - lse;
    acc += expf(lp) * lp;
    pout[(size_t)i * ITERS] = x;
  }
  acc = blockRedSum(acc, red);
  if (threadIdx.x == 0){
    float ne = -acc / logf((float)ODIM);
    float* c = d_out + (size_t)BATCH * ODIM * ITERS + (size_t)b * (2 * ITERS) + t;
    c[0] = ne; c[ITERS] = 1.f - ne;
  }
}

// ---------------- workspace layout (float offsets) ----------------
#define WS_KP     0u                      //  4096*512
#define WS_VP     2097152u                //  4096*512
#define WS_RING   4194304u                //  25*64*2048
#define WS_KV     7471104u                //  4096*512
#define WS_FEATS  9568256u                //  4096*12
#define WS_SYNCA  9617408u                //  64*512
#define WS_Q      9650176u
#define WS_QH     9682944u
#define WS_O      9715712u
#define WS_PRE    9748480u                //  64*2560
#define WS_SYN    9912320u                //  64*4096
#define WS_ACT    10174464u               //  64*2048
#define WS_SYNCO  10305536u               //  64*512
#define WS_PRED   10338304u               //  64*4096
#define WS_AA     10600448u
#define WS_BA     10633216u
#define WS_AO     10665984u
#define WS_BO     10698752u

extern "C" void kernel_launch(void* const* d_in, const int* in_sizes, int n_in,
                              void* d_out, int out_size, void* d_ws, size_t ws_size,
                              hipStream_t stream){
  const float* x        = (const float*)d_in[0];
  const float* kv_w     = (const float*)d_in[1];
  const float* kv_b     = (const float*)d_in[2];
  const float* kv_ln_g  = (const float*)d_in[3];
  const float* kv_ln_b  = (const float*)d_in[4];
  const float* q_w      = (const float*)d_in[5];
  const float* q_b      = (const float*)d_in[6];
  const float* wq       = (const float*)d_in[7];
  const float* wk       = (const float*)d_in[8];
  const float* wv       = (const float*)d_in[9];
  const float* wo       = (const float*)d_in[10];
  const float* bq       = (const float*)d_in[11];
  const float* bk       = (const float*)d_in[12];
  const float* bv       = (const float*)d_in[13];
  const float* bo       = (const float*)d_in[14];
  const float* syn_w    = (const float*)d_in[15];
  const float* syn_b    = (const float*)d_in[16];
  const float* syn_ln_g = (const float*)d_in[17];
  const float* syn_ln_b = (const float*)d_in[18];
  const float* nlm1_w   = (const float*)d_in[19];
  const float* nlm1_b   = (const float*)d_in[20];
  const float* nlm2_w   = (const float*)d_in[21];
  const float* nlm2_b   = (const float*)d_in[22];
  const float* start_act= (const float*)d_in[23];
  const float* start_tr = (const float*)d_in[24];
  const float* dec_a    = (const float*)d_in[25];
  const float* dec_o    = (const float*)d_in[26];
  const float* out_w    = (const float*)d_in[27];
  const float* out_b    = (const float*)d_in[28];
  const int*   idx_al   = (const int*)d_in[29];
  const int*   idx_ar   = (const int*)d_in[30];
  const int*   idx_ol   = (const int*)d_in[31];
  const int*   idx_or_  = (const int*)d_in[32];

  float* ws  = (float*)d_ws;
  float* out = (float*)d_out;

  float* pKP  = ws + WS_KP;   float* pVP   = ws + WS_VP;   float* pRING = ws + WS_RING;
  float* pKV  = ws + WS_KV;   float* pFT   = ws + WS_FEATS;
  float* pSA  = ws + WS_SYNCA;float* pQ    = ws + WS_Q;    float* pQH = ws + WS_QH;
  float* pO   = ws + WS_O;    float* pPRE  = ws + WS_PRE;  float* pSYN = ws + WS_SYN;
  float* pACT = ws + WS_ACT;  float* pSO   = ws + WS_SYNCO;float* pPRED = ws + WS_PRED;
  float* pAA  = ws + WS_AA;   float* pBA   = ws + WS_BA;   float* pAO = ws + WS_AO;
  float* pBO  = ws + WS_BO;

  auto gemm = [&](const float* A, int lda, const float* B, int ldb, float* C, int ldc,
                  const float* bias, int M, int N, int K){
    int total = (M / 16) * (N / 64);
    gemm_wmma_f32<<<(total + 7) / 8, 256, 0, stream>>>(A, lda, B, ldb, C, ldc, bias, M, N, K);
  };

  // ---- precompute ----
  init_kernel<<<512, 256, 0, stream>>>(start_act, pAA, pBA, pAO, pBO, pACT, pPRE);
  feats_kernel<<<192, 256, 0, stream>>>(x, pFT);
  gemm(pFT, 12, kv_w, 512, pKV, 512, kv_b, 4096, 512, 12);
  ln512_kernel<<<4096, 256, 0, stream>>>(pKV, kv_ln_g, kv_ln_b);
  gemm(pKV, 512, wk, 512, pKP, 512, bk, 4096, 512, 512);
  gemm(pKV, 512, wv, 512, pVP, 512, bv, 4096, 512, 512);

  // ---- 50 sequential iterations ----
  for (int t = 0; t < ITERS; t++){
    sync_kernel<<<128, 256, 0, stream>>>(pACT, idx_al, idx_ar, dec_a, pAA, pBA, pSA);
    gemm(pSA, 512, q_w, 512, pQ, 512, q_b, BATCH, 512, 512);
    gemm(pQ, 512, wq, 512, pQH, 512, bq, BATCH, 512, 512);
    attn_kernel<<<64, 256, 0, stream>>>(pQH, pKP, pVP, pO);
    gemm(pO, 512, wo, 512, pPRE, 2560, bo, BATCH, 512, 512);        // pre[:, 0:512]
    gemm(pPRE, 2560, syn_w, 4096, pSYN, 4096, syn_b, BATCH, 4096, 2560);
    glu_ln_kernel<<<64, 256, 0, stream>>>(pSYN, syn_ln_g, syn_ln_b, pRING, t);
    nlm_kernel<<<1024, 256, 0, stream>>>(pRING, start_tr, nlm1_w, nlm1_b,
                                         nlm2_w, nlm2_b, pACT, pPRE, t);
    sync_kernel<<<128, 256, 0, stream>>>(pACT, idx_ol, idx_or_, dec_o, pAO, pBO, pSO);
    gemm(pSO, 512, out_w, 4096, pPRED, 4096, out_b, BATCH, 4096, 512);
    finalize_kernel<<<64, 256, 0, stream>>>(pPRED, out, t);
  }
  (void)in_sizes; (void)n_in; (void)out_size; (void)ws_size;
}